// GlobalModel_29884382446070
// MI455X (gfx1250) — compile-verified
//
#include <hip/hip_runtime.h>

// CDNA5 / gfx1250. Wave32. Memory-bound streaming MLP chain:
//   128 MB total traffic @ 23.3 TB/s => ~5.5 us floor; compute (~0.7 GFLOP) is free.
// Strategy: coalesced B64 NT loads/stores, grid-stride, prefetch next chunk,
// and the 2x2 linear chain evaluated via V_WMMA_F32_16X16X4_F32 (Y = W*X,
// X as a 2x16 column batch; D rows 0..1 feed directly as next B).

typedef __attribute__((ext_vector_type(2))) float v2f;
typedef __attribute__((ext_vector_type(8))) float v8f;

__device__ __forceinline__ v8f wmma4(v2f a, v2f b, v8f c) {
    // 8-arg pattern: (neg_a, A, neg_b, B, c_mod, C, reuse_a, reuse_b)
    return __builtin_amdgcn_wmma_f32_16x16x4_f32(
        false, a, false, b, (short)0, c, false, false);
}

// Weight (torch [out,in], row-major flat w[0..3]) -> WMMA A layout:
// A VGPR0 lane m = W[m][0]; A VGPR1 lane m = W[m][1]; rows>=2 and K=2,3 zero.
__device__ __forceinline__ v2f makeA(const float* __restrict__ w, int lane) {
    v2f a;
    a[0] = (lane == 0) ? w[0] : ((lane == 1) ? w[2] : 0.0f);
    a[1] = (lane == 0) ? w[1] : ((lane == 1) ? w[3] : 0.0f);
    return a;
}

struct Net {
    v2f   A[8];          // w1_1,w1_2,w1_3,w2_1,w2_2,w3_1,w3_2,w4_1 in A layout
    float b0[8], b1[8];  // biases (wave-uniform -> SGPRs)
};

// One linear+bias+relu on activations held as (r0,r1) = B rows K=0,K=1.
__device__ __forceinline__ void layer_relu(const Net& n, int i,
                                           float& r0, float& r1, v8f cz) {
    v2f b; b[0] = r0; b[1] = r1;
    v8f d = wmma4(n.A[i], b, cz);
    r0 = fmaxf(d[0] + n.b0[i], 0.0f);
    r1 = fmaxf(d[1] + n.b1[i], 0.0f);
}

// Full reference network for one 16-column tile.
__device__ __forceinline__ void run_chain(const Net& n, float x0, float x1,
                                          float& f0, float& f1) {
    v8f cz = {};
    float r0 = x0, r1 = x1;
    layer_relu(n, 0, r0, r1, cz);           // out1 = relu(L1_1 x)
    layer_relu(n, 1, r0, r1, cz);           // a    = relu(L1_2 out1)
    layer_relu(n, 2, r0, r1, cz);           // relu(L1_3 a)
    float bb0 = r0 + x0, bb1 = r1 + x1;     // b = ... + x

    float t0 = bb0, t1 = bb1;
    layer_relu(n, 3, t0, t1, cz);           // t2  = relu(L2_1 b)
    layer_relu(n, 4, t0, t1, cz);           // relu(L2_2 t2)
    float o0 = t0 + bb0, o1 = t1 + bb1;     // out = ... + b

    float u0 = bb0, u1 = bb1;
    layer_relu(n, 5, u0, u1, cz);           // t3   = relu(L3_1 b)
    layer_relu(n, 6, u0, u1, cz);           // relu(L3_2 t3)
    float q0 = u0 + bb0, q1 = u1 + bb1;     // out3 = ... + b

    // final = relu(L4(out) + L4(out3)) ; each L4 carries +b4 -> 2*b4 total
    v2f bo; bo[0] = o0; bo[1] = o1;
    v8f d1 = wmma4(n.A[7], bo, cz);
    v2f bq; bq[0] = q0; bq[1] = q1;
    v8f d2 = wmma4(n.A[7], bq, cz);
    f0 = fmaxf(d1[0] + d2[0] + 2.0f * n.b0[7], 0.0f);
    f1 = fmaxf(d1[1] + d2[1] + 2.0f * n.b1[7], 0.0f);
}

__global__ __launch_bounds__(256) void fused_mlp_kernel(
    const float* __restrict__ x, float* __restrict__ out, long long nrows,
    const float* __restrict__ w11, const float* __restrict__ b11,
    const float* __restrict__ w12, const float* __restrict__ b12,
    const float* __restrict__ w13, const float* __restrict__ b13,
    const float* __restrict__ w21, const float* __restrict__ b21,
    const float* __restrict__ w22, const float* __restrict__ b22,
    const float* __restrict__ w31, const float* __restrict__ b31,
    const float* __restrict__ w32, const float* __restrict__ b32,
    const float* __restrict__ w41, const float* __restrict__ b41) {
    const int lane = (int)(threadIdx.x & 31u);
    const bool lo = lane < 16;

    Net n;
    n.A[0] = makeA(w11, lane); n.b0[0] = b11[0]; n.b1[0] = b11[1];
    n.A[1] = makeA(w12, lane); n.b0[1] = b12[0]; n.b1[1] = b12[1];
    n.A[2] = makeA(w13, lane); n.b0[2] = b13[0]; n.b1[2] = b13[1];
    n.A[3] = makeA(w21, lane); n.b0[3] = b21[0]; n.b1[3] = b21[1];
    n.A[4] = makeA(w22, lane); n.b0[4] = b22[0]; n.b1[4] = b22[1];
    n.A[5] = makeA(w31, lane); n.b0[5] = b31[0]; n.b1[5] = b31[1];
    n.A[6] = makeA(w32, lane); n.b0[6] = b32[0]; n.b1[6] = b32[1];
    n.A[7] = makeA(w41, lane); n.b0[7] = b41[0]; n.b1[7] = b41[1];

    const v2f* __restrict__ xp = (const v2f*)x;
    v2f* __restrict__ op = (v2f*)out;

    const long long wave   = (long long)blockIdx.x * (blockDim.x >> 5) + (threadIdx.x >> 5);
    const long long nwaves = (long long)gridDim.x * (blockDim.x >> 5);
    const long long nchunks = (nrows + 31) / 32;   // 32 rows / wave / chunk

    for (long long ch = wave; ch < nchunks; ch += nwaves) {
        const long long row = ch * 32 + lane;

        v2f xv = {};
        if (row < nrows) xv = __builtin_nontemporal_load(xp + row);  // coalesced B64, NT (no reuse)

        // Prefetch next chunk this wave will touch (global_prefetch_b8).
        const long long nch = ch + nwaves;
        if (nch < nchunks)
            __builtin_prefetch((const void*)(xp + (nch * 32 + lane)), 0, 1);

        // tile1 = rows 16..31 of the chunk, swapped into lanes 0..15
        const float sx0 = __shfl_xor(xv[0], 16);
        const float sx1 = __shfl_xor(xv[1], 16);

        // Upper half-lanes must stay finite (they feed B K=2,3 against A's zeros).
        const float t0x0 = lo ? xv[0] : 0.0f, t0x1 = lo ? xv[1] : 0.0f;
        const float t1x0 = lo ? sx0   : 0.0f, t1x1 = lo ? sx1   : 0.0f;

        float f00, f01, f10, f11;
        run_chain(n, t0x0, t0x1, f00, f01);   // two independent WMMA chains:
        run_chain(n, t1x0, t1x1, f10, f11);   // scheduler hides D->B hazards

        // Recombine: lanes 0..15 store tile0, lanes 16..31 store tile1.
        const float g0 = __shfl_xor(f10, 16);
        const float g1 = __shfl_xor(f11, 16);
        v2f ov;
        ov[0] = lo ? f00 : g0;
        ov[1] = lo ? f01 : g1;
        if (row < nrows) __builtin_nontemporal_store(ov, op + row);
    }
}

extern "C" void kernel_launch(void* const* d_in, const int* in_sizes, int n_in,
                              void* d_out, int out_size, void* d_ws, size_t ws_size,
                              hipStream_t stream) {
    const float* x = (const float*)d_in[0];
    const long long nrows = (long long)in_sizes[0] / 2;
    const long long nchunks = (nrows + 31) / 32;
    const int waves_per_block = 256 / 32;
    long long blocks = (nchunks + waves_per_block - 1) / waves_per_block;
    if (blocks > 8192) blocks = 8192;   // 65536 waves in flight; grid-stride covers rest
    if (blocks < 1) blocks = 1;

    fused_mlp_kernel<<<(int)blocks, 256, 0, stream>>>(
        x, (float*)d_out, nrows,
        (const float*)d_in[1],  (const float*)d_in[2],
        (const float*)d_in[3],  (const float*)d_in[4],
        (const float*)d_in[5],  (const float*)d_in[6],
        (const float*)d_in[7],  (const float*)d_in[8],
        (const float*)d_in[9],  (const float*)d_in[10],
        (const float*)d_in[11], (const float*)d_in[12],
        (const float*)d_in[13], (const float*)d_in[14],
        (const float*)d_in[15], (const float*)d_in[16]);
}